// Model_58866821759238
// MI455X (gfx1250) — compile-verified
//
#include <hip/hip_runtime.h>
#include <cstddef>

typedef __attribute__((ext_vector_type(2))) float v2f;
typedef __attribute__((ext_vector_type(8))) float v8f;

#define C_DIM 64
#define F_DIM 384

// ---------------------------------------------------------------------------
// Native fp32 WMMA: D(16x16) = A(16x4) * B(4x16) + C
// args: (neg_a, A, neg_b, B, c_mod, C, reuse_a, reuse_b)
// ---------------------------------------------------------------------------
__device__ __forceinline__ v8f wmma4(v2f a, v2f b, v8f c) {
  return __builtin_amdgcn_wmma_f32_16x16x4_f32(false, a, false, b, (short)0, c,
                                               false, false);
}

// ---------------------------------------------------------------------------
// out[N,64] = X[N,384] @ W[64,384]^T + bias
// One wave owns a 16-row stripe and computes ALL FOUR 16-col tiles, so each
// A element is loaded exactly once (non-temporal: X is streamed one time and
// must not evict the L2-resident graph buffers).
// ---------------------------------------------------------------------------
__global__ __launch_bounds__(128) void proj_wmma(
    const float* __restrict__ X, const float* __restrict__ W,
    const float* __restrict__ bias, float* __restrict__ out, int n) {
  const int lane = threadIdx.x & 31;
  const int wave = threadIdx.x >> 5;
  const int lm   = lane & 15;   // A row within stripe / B,C,D column
  const int half = lane >> 4;   // selects K pair
  const int rowTiles = (n + 15) >> 4;
  const int rt = blockIdx.x * 4 + wave;
  if (rt >= rowTiles) return;   // wave-uniform exit
  const int r0 = rt << 4;
  const int row = r0 + lm;
  const int ar = (row < n) ? row : 0;  // redirect tail rows (their D rows unstored)
  const float* xr = X + (size_t)ar * F_DIM;
  const float* w0 = W + (size_t)lm * F_DIM;  // col-tile t uses w0 + t*16*F_DIM

  v8f acc0 = {}, acc1 = {}, acc2 = {}, acc3 = {};
  for (int k = half * 2; k < F_DIM; k += 4) {
    v2f a = __builtin_nontemporal_load((const v2f*)(xr + k));
    v2f b0 = *(const v2f*)(w0 + 0 * 16 * F_DIM + k);
    v2f b1 = *(const v2f*)(w0 + 1 * 16 * F_DIM + k);
    v2f b2 = *(const v2f*)(w0 + 2 * 16 * F_DIM + k);
    v2f b3 = *(const v2f*)(w0 + 3 * 16 * F_DIM + k);
    acc0 = wmma4(a, b0, acc0);
    acc1 = wmma4(a, b1, acc1);
    acc2 = wmma4(a, b2, acc2);
    acc3 = wmma4(a, b3, acc3);
  }
  const float bb0 = bias[lm], bb1 = bias[lm + 16];
  const float bb2 = bias[lm + 32], bb3 = bias[lm + 48];
  if (r0 + 16 <= n) {  // full tile: unguarded stores (dominant path)
#pragma unroll
    for (int v = 0; v < 8; ++v) {
      float* o = out + (size_t)(r0 + v + 8 * half) * C_DIM + lm;
      o[0] = acc0[v] + bb0;  o[16] = acc1[v] + bb1;
      o[32] = acc2[v] + bb2; o[48] = acc3[v] + bb3;
    }
  } else {
#pragma unroll
    for (int v = 0; v < 8; ++v) {
      const int rr = r0 + v + 8 * half;
      if (rr < n) {
        float* o = out + (size_t)rr * C_DIM + lm;
        o[0] = acc0[v] + bb0;  o[16] = acc1[v] + bb1;
        o[32] = acc2[v] + bb2; o[48] = acc3[v] + bb3;
      }
    }
  }
}

// ---------------------------------------------------------------------------
// SAGEConv dense part, all four col tiles per wave:
//   out[n,64] += (agg/max(cnt,1)) @ Wl^T + bl + xdst @ Wr^T
// 8 WMMAs per k-step across 4 independent accumulators.
// ---------------------------------------------------------------------------
__global__ __launch_bounds__(128) void sage_gemm(
    const float* __restrict__ agg, const float* __restrict__ cnt,
    const float* __restrict__ xdst, const float* __restrict__ Wl,
    const float* __restrict__ bl, const float* __restrict__ Wr,
    float* __restrict__ out, int n) {
  const int lane = threadIdx.x & 31;
  const int wave = threadIdx.x >> 5;
  const int lm   = lane & 15;
  const int half = lane >> 4;
  const int rowTiles = (n + 15) >> 4;
  const int rt = blockIdx.x * 4 + wave;
  if (rt >= rowTiles) return;
  const int r0 = rt << 4;
  const int row = r0 + lm;
  const int ar  = (row < n) ? row : 0;
  const float scl = 1.0f / fmaxf(cnt[ar], 1.0f);
  const float* agr = agg  + (size_t)ar * C_DIM;
  const float* xr  = xdst + (size_t)ar * C_DIM;
  const float* wl0 = Wl + (size_t)lm * C_DIM;
  const float* wr0 = Wr + (size_t)lm * C_DIM;

  v8f acc0 = {}, acc1 = {}, acc2 = {}, acc3 = {};
  for (int k = half * 2; k < C_DIM; k += 4) {
    v2f am = *(const v2f*)(agr + k);
    am.x *= scl; am.y *= scl;                 // mean = agg / max(cnt,1)
    v2f ax = *(const v2f*)(xr + k);
    v2f bl0 = *(const v2f*)(wl0 + 0 * 16 * C_DIM + k);
    v2f bl1 = *(const v2f*)(wl0 + 1 * 16 * C_DIM + k);
    v2f bl2 = *(const v2f*)(wl0 + 2 * 16 * C_DIM + k);
    v2f bl3 = *(const v2f*)(wl0 + 3 * 16 * C_DIM + k);
    v2f br0 = *(const v2f*)(wr0 + 0 * 16 * C_DIM + k);
    v2f br1 = *(const v2f*)(wr0 + 1 * 16 * C_DIM + k);
    v2f br2 = *(const v2f*)(wr0 + 2 * 16 * C_DIM + k);
    v2f br3 = *(const v2f*)(wr0 + 3 * 16 * C_DIM + k);
    acc0 = wmma4(am, bl0, acc0); acc0 = wmma4(ax, br0, acc0);
    acc1 = wmma4(am, bl1, acc1); acc1 = wmma4(ax, br1, acc1);
    acc2 = wmma4(am, bl2, acc2); acc2 = wmma4(ax, br2, acc2);
    acc3 = wmma4(am, bl3, acc3); acc3 = wmma4(ax, br3, acc3);
  }
  const float bb0 = bl[lm], bb1 = bl[lm + 16];
  const float bb2 = bl[lm + 32], bb3 = bl[lm + 48];
  if (r0 + 16 <= n) {
#pragma unroll
    for (int v = 0; v < 8; ++v) {
      float* o = out + (size_t)(r0 + v + 8 * half) * C_DIM + lm;
      o[0] += acc0[v] + bb0;  o[16] += acc1[v] + bb1;
      o[32] += acc2[v] + bb2; o[48] += acc3[v] + bb3;
    }
  } else {
#pragma unroll
    for (int v = 0; v < 8; ++v) {
      const int rr = r0 + v + 8 * half;
      if (rr < n) {
        float* o = out + (size_t)rr * C_DIM + lm;
        o[0] += acc0[v] + bb0;  o[16] += acc1[v] + bb1;
        o[32] += acc2[v] + bb2; o[48] += acc3[v] + bb3;
      }
    }
  }
}

// ---------------------------------------------------------------------------
// Edge scatter: agg[dst] += x_src[src]; cnt[dst] += 1.  One wave per edge,
// each lane carries 2 channels (float2 coalesced); fp32 global atomics (L2).
// ---------------------------------------------------------------------------
__global__ __launch_bounds__(256) void scatter_mean(
    const float* __restrict__ xsrc, const int* __restrict__ srcIdx,
    const int* __restrict__ dstIdx, int E, float* __restrict__ agg,
    float* __restrict__ cnt) {
  const int lane = threadIdx.x & 31;
  int w  = (int)((blockIdx.x * blockDim.x + threadIdx.x) >> 5);
  int nw = (int)((gridDim.x * blockDim.x) >> 5);
  for (int e = w; e < E; e += nw) {
    const int s = srcIdx[e];
    const int d = dstIdx[e];
    const float2 v = ((const float2*)(xsrc + (size_t)s * C_DIM))[lane];
    float* ap = agg + (size_t)d * C_DIM + lane * 2;
    atomicAdd(ap,     v.x);
    atomicAdd(ap + 1, v.y);
    if (lane == 0) atomicAdd(cnt + d, 1.0f);
  }
}

// ---------------------------------------------------------------------------
// score[e] = dot(zu[eli0[e]], zs[eli1[e]])  — wave per edge, shfl reduction
// ---------------------------------------------------------------------------
__global__ __launch_bounds__(256) void score_kernel(
    const float* __restrict__ zu, const float* __restrict__ zs,
    const int* __restrict__ ui, const int* __restrict__ si, int E,
    float* __restrict__ out) {
  const int lane = threadIdx.x & 31;
  int w  = (int)((blockIdx.x * blockDim.x + threadIdx.x) >> 5);
  int nw = (int)((gridDim.x * blockDim.x) >> 5);
  for (int e = w; e < E; e += nw) {
    const int u = ui[e], s = si[e];
    const float2 a = ((const float2*)(zu + (size_t)u * C_DIM))[lane];
    const float2 b = ((const float2*)(zs + (size_t)s * C_DIM))[lane];
    float p = a.x * b.x + a.y * b.y;
    for (int o = 16; o > 0; o >>= 1) p += __shfl_xor(p, o, 32);
    if (lane == 0) out[e] = p;
  }
}

__global__ void zero_kernel(float* __restrict__ p, size_t n) {
  size_t i  = (size_t)blockIdx.x * blockDim.x + threadIdx.x;
  size_t st = (size_t)gridDim.x * blockDim.x;
  for (; i < n; i += st) p[i] = 0.0f;
}

__global__ void relu_kernel(float* __restrict__ p, size_t n) {
  size_t i  = (size_t)blockIdx.x * blockDim.x + threadIdx.x;
  size_t st = (size_t)gridDim.x * blockDim.x;
  for (; i < n; i += st) p[i] = fmaxf(p[i], 0.0f);
}

static inline int iminc(long a, long b) { return (int)(a < b ? a : b); }

extern "C" void kernel_launch(void* const* d_in, const int* in_sizes, int n_in,
                              void* d_out, int out_size, void* d_ws,
                              size_t ws_size, hipStream_t stream) {
  const int NU = 100000, NS = 50000, NG = 50, NW = 20000, NT = 10;

  const float* reviews  = (const float*)d_in[0];
  const float* overview = (const float*)d_in[1];
  const float* g_emb    = (const float*)d_in[2];
  const float* w_emb    = (const float*)d_in[3];
  const float* t_emb    = (const float*)d_in[4];
  const float* Wu  = (const float*)d_in[5];
  const float* bu  = (const float*)d_in[6];
  const float* Ws_ = (const float*)d_in[7];
  const float* bs  = (const float*)d_in[8];
  const float* Wl1 = (const float*)d_in[9];
  const float* bl1 = (const float*)d_in[10];
  const float* Wr1 = (const float*)d_in[11];
  const float* Wl2 = (const float*)d_in[12];
  const float* bl2 = (const float*)d_in[13];
  const float* Wr2 = (const float*)d_in[14];
  const int* e_u2s = (const int*)d_in[15];
  const int* e_g2s = (const int*)d_in[16];
  const int* e_w2s = (const int*)d_in[17];
  const int* e_t2s = (const int*)d_in[18];
  const int* eli   = (const int*)d_in[19];
  float* outp = (float*)d_out;

  const int Eu = in_sizes[15] / 2, Eg = in_sizes[16] / 2;
  const int Ew = in_sizes[17] / 2, Et = in_sizes[18] / 2;
  const int El = in_sizes[19] / 2;

  // ---- workspace layout (fp32) ----
  float* w = (float*)d_ws;
  float* p_u = w;  w += (size_t)NU * C_DIM;   // layer-0 users (later: layer-2 users)
  float* p_s = w;  w += (size_t)NS * C_DIM;   // layer-0 series (later: layer-2 series)
  float* yu  = w;  w += (size_t)NU * C_DIM;   // layer-1 outputs (contiguous block)
  float* ys  = w;  w += (size_t)NS * C_DIM;
  float* yg  = w;  w += (size_t)NG * C_DIM;
  float* yw_ = w;  w += (size_t)NW * C_DIM;
  float* yt  = w;  w += (size_t)NT * C_DIM;
  float* agg = w;  w += (size_t)NU * C_DIM;   // per-conv scatter accumulator
  float* cnt = w;  w += (size_t)NU;

  // ---- input projections (WMMA fp32) ----
  {
    int rtU = (NU + 15) / 16, rtS = (NS + 15) / 16;
    proj_wmma<<<(rtU + 3) / 4, 128, 0, stream>>>(reviews, Wu, bu, p_u, NU);
    proj_wmma<<<(rtS + 3) / 4, 128, 0, stream>>>(overview, Ws_, bs, p_s, NS);
  }

  // zero all layer-1 outputs (contiguous region starting at yu)
  const size_t yTot = (size_t)(NU + NS + NG + NW + NT) * C_DIM;
  zero_kernel<<<2048, 256, 0, stream>>>(yu, yTot);

  auto conv = [&](const float* xsrc, const float* xdst, const int* s,
                  const int* d, int E, int nDst, const float* Wl,
                  const float* bl, const float* Wr, float* ob) {
    const size_t na = (size_t)nDst * C_DIM;
    zero_kernel<<<iminc(2048, (long)((na + 255) / 256)), 256, 0, stream>>>(agg, na);
    zero_kernel<<<iminc(256, (long)((nDst + 255) / 256)), 256, 0, stream>>>(cnt, (size_t)nDst);
    scatter_mean<<<iminc(2048, ((long)E + 7) / 8), 256, 0, stream>>>(xsrc, s, d, E, agg, cnt);
    const int rowTiles = (nDst + 15) / 16;
    sage_gemm<<<(rowTiles + 3) / 4, 128, 0, stream>>>(agg, cnt, xdst, Wl, bl, Wr, ob, nDst);
  };

  const int WSZ = C_DIM * C_DIM;  // 4096 floats per [64,64] weight

  // ---- layer 1: all 8 edge types ----
  conv(p_u,   p_s,   e_u2s,      e_u2s + Eu, Eu, NS, Wl1 + 0 * WSZ, bl1 + 0 * C_DIM, Wr1 + 0 * WSZ, ys);
  conv(p_s,   p_u,   e_u2s + Eu, e_u2s,      Eu, NU, Wl1 + 1 * WSZ, bl1 + 1 * C_DIM, Wr1 + 1 * WSZ, yu);
  conv(g_emb, p_s,   e_g2s,      e_g2s + Eg, Eg, NS, Wl1 + 2 * WSZ, bl1 + 2 * C_DIM, Wr1 + 2 * WSZ, ys);
  conv(p_s,   g_emb, e_g2s + Eg, e_g2s,      Eg, NG, Wl1 + 3 * WSZ, bl1 + 3 * C_DIM, Wr1 + 3 * WSZ, yg);
  conv(w_emb, p_s,   e_w2s,      e_w2s + Ew, Ew, NS, Wl1 + 4 * WSZ, bl1 + 4 * C_DIM, Wr1 + 4 * WSZ, ys);
  conv(p_s,   w_emb, e_w2s + Ew, e_w2s,      Ew, NW, Wl1 + 5 * WSZ, bl1 + 5 * C_DIM, Wr1 + 5 * WSZ, yw_);
  conv(t_emb, p_s,   e_t2s,      e_t2s + Et, Et, NS, Wl1 + 6 * WSZ, bl1 + 6 * C_DIM, Wr1 + 6 * WSZ, ys);
  conv(p_s,   t_emb, e_t2s + Et, e_t2s,      Et, NT, Wl1 + 7 * WSZ, bl1 + 7 * C_DIM, Wr1 + 7 * WSZ, yt);

  // ReLU over all layer-1 outputs
  relu_kernel<<<2048, 256, 0, stream>>>(yu, yTot);

  // ---- layer 2: only convs whose dst is users/series feeds the scorer ----
  // reuse p_u/p_s (contiguous) as layer-2 outputs
  zero_kernel<<<2048, 256, 0, stream>>>(p_u, (size_t)(NU + NS) * C_DIM);

  conv(yu,  ys, e_u2s,      e_u2s + Eu, Eu, NS, Wl2 + 0 * WSZ, bl2 + 0 * C_DIM, Wr2 + 0 * WSZ, p_s);
  conv(ys,  yu, e_u2s + Eu, e_u2s,      Eu, NU, Wl2 + 1 * WSZ, bl2 + 1 * C_DIM, Wr2 + 1 * WSZ, p_u);
  conv(yg,  ys, e_g2s,      e_g2s + Eg, Eg, NS, Wl2 + 2 * WSZ, bl2 + 2 * C_DIM, Wr2 + 2 * WSZ, p_s);
  conv(yw_, ys, e_w2s,      e_w2s + Ew, Ew, NS, Wl2 + 4 * WSZ, bl2 + 4 * C_DIM, Wr2 + 4 * WSZ, p_s);
  conv(yt,  ys, e_t2s,      e_t2s + Et, Et, NS, Wl2 + 6 * WSZ, bl2 + 6 * C_DIM, Wr2 + 6 * WSZ, p_s);

  // ---- dot-product scoring on edge_label_index ----
  score_kernel<<<iminc(2048, ((long)El + 7) / 8), 256, 0, stream>>>(
      p_u, p_s, eli, eli + El, El, outp);
}